// Net_39015482917231
// MI455X (gfx1250) — compile-verified
//
#include <hip/hip_runtime.h>
#include <math.h>
#include <stdint.h>

#define NN 6890
#define NP 6912      // NN padded to multiple of 256 (27*256), also 54*128
#define EE 55120
#define KK 5
#define K3T 125
#define KCH 25       // kernel-index chunk (5 chunks of 25)
#define NCHUNK 5
#define FC2P 6912    // fc2 column count padded (54*128)
#define LDH64 1664   // KCH*64=1600 padded to 13*128
#define LDH32 800    // layer-1 (KCH*32), simple-kernel path (multiple of 32)

// LDS strides (dwords) chosen for conflict-free WMMA fragment reads
#define ASTRIDE 36   // 128 x 32 A tile: 16 distinct banks over 16 rows, b64/b128 aligned
#define BSTRIDE 144  // 32 x 128 B tile: 144 mod 64 = 16 -> row pairs in disjoint bank groups

typedef float v2f __attribute__((ext_vector_type(2)));
typedef float v8f __attribute__((ext_vector_type(8)));

static inline int cdiv(int a, int b) { return (a + b - 1) / b; }

// ---------------- utility kernels ----------------

__global__ void zero_kernel(float* p, int n) {
    int i = blockIdx.x * blockDim.x + threadIdx.x;
    if (i < n) p[i] = 0.f;
}

__global__ void padmat_kernel(const float* __restrict__ src, float* __restrict__ dst,
                              int rows, int cols, int ldsrc,
                              int rowsPad, int colsPad, int lddst) {
    int i = blockIdx.x * blockDim.x + threadIdx.x;
    int tot = rowsPad * colsPad;
    if (i >= tot) return;
    int c = i % colsPad, r = i / colsPad;
    dst[(size_t)r * lddst + c] = (r < rows && c < cols) ? src[(size_t)r * ldsrc + c] : 0.f;
}

__global__ void basis_kernel(const float* __restrict__ pseudo,
                             float* __restrict__ basis, int* __restrict__ flat) {
    int e = blockIdx.x * blockDim.x + threadIdx.x;
    if (e >= EE) return;
    float frac[3]; int bot[3];
    for (int d = 0; d < 3; d++) {
        float p = pseudo[e * 3 + d] * (float)(KK - 1);
        float b = floorf(p);
        b = fminf(fmaxf(b, 0.f), (float)(KK - 2));
        bot[d] = (int)b;
        frac[d] = p - b;
    }
    const int str[3] = {1, KK, KK * KK};
    for (int b = 0; b < 8; b++) {
        float w = 1.f; int f = 0;
        for (int d = 0; d < 3; d++) {
            int bit = (b >> d) & 1;
            w *= bit ? frac[d] : (1.f - frac[d]);
            f += (bot[d] + bit) * str[d];
        }
        basis[e * 8 + b] = w;
        flat[e * 8 + b] = f;
    }
}

__global__ void deg_kernel(const int* __restrict__ dst, float* __restrict__ deg) {
    int e = blockIdx.x * blockDim.x + threadIdx.x;
    if (e < EE) atomicAdd(&deg[dst[e]], 1.f);
}

__global__ void pack_kernel(const float* __restrict__ Wc, float* __restrict__ Wt,
                            int cin, int cout, int k0, int kn, int kpad, int ldw) {
    int i = blockIdx.x * blockDim.x + threadIdx.x;
    int tot = kpad * ldw;
    if (i >= tot) return;
    int col = i % ldw;
    int ci = i / ldw;
    int kk = col / cout;
    int o = col % cout;
    float v = (ci < cin && kk < kn) ? Wc[((k0 + kk) * cin + ci) * cout + o] : 0.f;
    Wt[(size_t)ci * ldw + col] = v;
}

// ---------------- shared store epilogue ----------------

__device__ __forceinline__ void store_tile(v8f c, float* __restrict__ C, int ldc,
                                           int r0, int c0, int l15, int hi,
                                           int Ms, int Ns,
                                           const float* __restrict__ bias, int act) {
    int col = c0 + l15;
    if (col >= Ns) return;
    float bv = (bias != nullptr) ? bias[col] : 0.f;
    int rbase = r0 + (hi << 3);
    for (int j = 0; j < 8; j++) {
        int r = rbase + j;
        if (r < Ms) {
            float v = c[j] + bv;
            if (act == 1) v = v > 0.f ? v : (expf(v) - 1.f);
            C[(size_t)r * ldc + col] = v;
        }
    }
}

// ---------------- simple fp32 WMMA GEMM (global operands) ----------------
// Used for small / K-odd GEMMs (root terms, layer-1 K=4). Mp%32==0, Np%32==0, K%4==0.
__global__ void gemm_wmma_f32(const float* __restrict__ A, const float* __restrict__ B,
                              float* __restrict__ C, int Mp, int Np, int K,
                              int lda, int ldb, int ldc, int Ms, int Ns,
                              const float* __restrict__ bias, int act) {
    const int lane = threadIdx.x & 31;
    const int wavesPerBlock = blockDim.x >> 5;
    const int wave = blockIdx.x * wavesPerBlock + (threadIdx.x >> 5);
    const int tm = Mp >> 5, tn = Np >> 5;
    const int tiles = tm * tn;
    const int nwaves = gridDim.x * wavesPerBlock;
    const int l15 = lane & 15;
    const int hi = lane >> 4;
    const int koff = hi << 1;

    for (int t = wave; t < tiles; t += nwaves) {
        const int tmi = t / tn, tni = t % tn;
        const int row0 = tmi << 5, col0 = tni << 5;
        v8f c00 = {}, c01 = {}, c10 = {}, c11 = {};
        const float* Ar0 = A + (size_t)(row0 + l15) * lda + koff;
        const float* Ar1 = A + (size_t)(row0 + 16 + l15) * lda + koff;
        const float* Bc0 = B + col0 + l15;
        const float* Bc1 = Bc0 + 16;
        for (int k0 = 0; k0 < K; k0 += 4) {
            v2f a0 = *(const v2f*)(Ar0 + k0);
            v2f a1 = *(const v2f*)(Ar1 + k0);
            const size_t bo = (size_t)(k0 + koff) * ldb;
            v2f b0, b1;
            b0.x = Bc0[bo];       b0.y = Bc0[bo + ldb];
            b1.x = Bc1[bo];       b1.y = Bc1[bo + ldb];
            c00 = __builtin_amdgcn_wmma_f32_16x16x4_f32(false, a0, false, b0, (short)0, c00, false, false);
            c01 = __builtin_amdgcn_wmma_f32_16x16x4_f32(false, a0, false, b1, (short)0, c01, false, false);
            c10 = __builtin_amdgcn_wmma_f32_16x16x4_f32(false, a1, false, b0, (short)0, c10, false, false);
            c11 = __builtin_amdgcn_wmma_f32_16x16x4_f32(false, a1, false, b1, (short)0, c11, false, false);
        }
        store_tile(c00, C, ldc, row0,      col0,      l15, hi, Ms, Ns, bias, act);
        store_tile(c01, C, ldc, row0,      col0 + 16, l15, hi, Ms, Ns, bias, act);
        store_tile(c10, C, ldc, row0 + 16, col0,      l15, hi, Ms, Ns, bias, act);
        store_tile(c11, C, ldc, row0 + 16, col0 + 16, l15, hi, Ms, Ns, bias, act);
    }
}

// ---------------- LDS-staged fp32 WMMA GEMM, double-buffered async DMA ----------------
// Requires Mp%128==0, Np%128==0, K%32==0. Block = 256 threads / 8 waves, 128x128 tile;
// wave grid 4(M) x 2(N), each wave 32x64 = 8 WMMA accumulators. Panels staged with
// GLOBAL_LOAD_ASYNC_TO_LDS_B128; ASYNCcnt in-order retirement gives 1-deep pipeline:
// issue stage s+1, s_wait_asynccnt 8 -> stage s landed while s+1 still in flight.

__device__ __forceinline__ void async_b128(uint32_t lds_off, const float* gptr) {
    asm volatile("global_load_async_to_lds_b128 %0, %1, off"
                 :: "v"(lds_off), "v"((unsigned long long)(uintptr_t)gptr)
                 : "memory");
}

#define WAIT_ASYNC_0() asm volatile("s_wait_asynccnt 0x0" ::: "memory")
#define WAIT_ASYNC_8() asm volatile("s_wait_asynccnt 0x8" ::: "memory")

// issue one stage: A[128][32] and B[32][128] panels, 8 async b128 per thread
__device__ __forceinline__ void stage_panels(uint32_t asB, uint32_t bsB,
                                             const float* Ag, const float* Bg,
                                             int lda, int ldb, int t) {
    #pragma unroll
    for (int p = 0; p < 4; p++) {
        int idx = t + p * 256;
        int r = idx >> 3;
        int c = (idx & 7) << 2;
        async_b128(asB + (uint32_t)(r * ASTRIDE + c) * 4u, Ag + (size_t)r * lda + c);
    }
    #pragma unroll
    for (int p = 0; p < 4; p++) {
        int idx = t + p * 256;
        int r = idx >> 5;
        int c = (idx & 31) << 2;
        async_b128(bsB + (uint32_t)(r * BSTRIDE + c) * 4u, Bg + (size_t)r * ldb + c);
    }
}

__global__ void gemm_lds_wmma_f32(const float* __restrict__ A, const float* __restrict__ B,
                                  float* __restrict__ C, int K,
                                  int lda, int ldb, int ldc, int tnB,
                                  int Ms, int Ns,
                                  const float* __restrict__ bias, int act) {
    __shared__ float As[2][128 * ASTRIDE];
    __shared__ float Bs[2][32 * BSTRIDE];

    const int t = threadIdx.x;
    const int lane = t & 31;
    const int l15 = lane & 15;
    const int hi = lane >> 4;
    const int koff = hi << 1;
    const int w = t >> 5;
    const int wm = w >> 1;          // 0..3
    const int wn = w & 1;           // 0..1

    const int bm = blockIdx.x / tnB;
    const int bn = blockIdx.x % tnB;
    const int rowBase = bm << 7;
    const int colBase = bn << 7;

    const uint32_t asB[2] = {(uint32_t)(uintptr_t)&As[0][0], (uint32_t)(uintptr_t)&As[1][0]};
    const uint32_t bsB[2] = {(uint32_t)(uintptr_t)&Bs[0][0], (uint32_t)(uintptr_t)&Bs[1][0]};

    const float* Abase = A + (size_t)rowBase * lda;
    const float* Bbase = B + colBase;

    v8f acc[2][4];
    #pragma unroll
    for (int mi = 0; mi < 2; mi++)
        #pragma unroll
        for (int s = 0; s < 4; s++)
            acc[mi][s] = (v8f){};

    const int nStages = K >> 5;

    // prologue: stage 0 into buffer 0
    stage_panels(asB[0], bsB[0], Abase, Bbase, lda, ldb, t);

    for (int st = 0; st < nStages; st++) {
        const int cur = st & 1;
        if (st + 1 < nStages) {
            // overlap: issue next stage into the other buffer, then wait for current
            const int nxt = cur ^ 1;
            const int kc = (st + 1) << 5;
            stage_panels(asB[nxt], bsB[nxt], Abase + kc, Bbase + (size_t)kc * ldb,
                         lda, ldb, t);
            WAIT_ASYNC_8();
        } else {
            WAIT_ASYNC_0();
        }
        __syncthreads();

        const float* Ar0 = &As[cur][(wm * 32 + l15) * ASTRIDE + koff];
        const float* Ar1 = &As[cur][(wm * 32 + 16 + l15) * ASTRIDE + koff];
        const float* Bc  = &Bs[cur][wn * 64 + l15];
        #pragma unroll
        for (int kk = 0; kk < 32; kk += 4) {
            v2f a0 = *(const v2f*)(Ar0 + kk);
            v2f a1 = *(const v2f*)(Ar1 + kk);
            const int br0 = (kk + koff) * BSTRIDE;
            #pragma unroll
            for (int s = 0; s < 4; s++) {
                v2f b;
                b.x = Bc[br0 + s * 16];
                b.y = Bc[br0 + BSTRIDE + s * 16];
                acc[0][s] = __builtin_amdgcn_wmma_f32_16x16x4_f32(false, a0, false, b, (short)0, acc[0][s], false, false);
                acc[1][s] = __builtin_amdgcn_wmma_f32_16x16x4_f32(false, a1, false, b, (short)0, acc[1][s], false, false);
            }
        }
        __syncthreads();   // fences buffer reuse: stage st+2 overwrites buffer cur
    }

    const int row0 = rowBase + wm * 32;
    const int col0 = colBase + wn * 64;
    #pragma unroll
    for (int mi = 0; mi < 2; mi++)
        #pragma unroll
        for (int s = 0; s < 4; s++)
            store_tile(acc[mi][s], C, ldc, row0 + mi * 16, col0 + s * 16,
                       l15, hi, Ms, Ns, bias, act);
}

// ---------------- edge message + scatter ----------------
__global__ void edge_kernel(const float* __restrict__ h, const int* __restrict__ src,
                            const int* __restrict__ dst, const float* __restrict__ basis,
                            const int* __restrict__ flat, float* __restrict__ agg,
                            int cout, int k0, int k1, int ldh) {
    int tid = blockIdx.x * blockDim.x + threadIdx.x;
    int e = tid / cout;
    int o = tid % cout;
    if (e >= EE) return;
    int s = src[e];
    float acc = 0.f;
    bool any = false;
    for (int b = 0; b < 8; b++) {
        int k = flat[e * 8 + b];
        if (k >= k0 && k < k1) {
            acc += basis[e * 8 + b] * h[(size_t)s * ldh + (k - k0) * cout + o];
            any = true;
        }
    }
    if (any) atomicAdd(&agg[dst[e] * cout + o], acc);
}

__global__ void node_kernel(const float* __restrict__ agg, const float* __restrict__ rootb,
                            const float* __restrict__ bias, const float* __restrict__ deg,
                            float* __restrict__ xout, int cout, int ldx, int ldr) {
    int tid = blockIdx.x * blockDim.x + threadIdx.x;
    if (tid >= NN * cout) return;
    int n = tid / cout, o = tid % cout;
    float d = fmaxf(deg[n], 1.f);
    float v = agg[(size_t)n * cout + o] / d + rootb[(size_t)n * ldr + o] + bias[o];
    xout[(size_t)n * ldx + o] = v > 0.f ? v : (expf(v) - 1.f);
}

__global__ void logsoftmax_kernel(float* __restrict__ out, int C) {
    __shared__ float red[256];
    int row = blockIdx.x;
    float* p = out + (size_t)row * C;
    float mx = -3.402823466e+38f;
    for (int i = threadIdx.x; i < C; i += blockDim.x) mx = fmaxf(mx, p[i]);
    red[threadIdx.x] = mx;
    __syncthreads();
    for (int s = 128; s > 0; s >>= 1) {
        if (threadIdx.x < s) red[threadIdx.x] = fmaxf(red[threadIdx.x], red[threadIdx.x + s]);
        __syncthreads();
    }
    mx = red[0];
    __syncthreads();
    float sum = 0.f;
    for (int i = threadIdx.x; i < C; i += blockDim.x) sum += expf(p[i] - mx);
    red[threadIdx.x] = sum;
    __syncthreads();
    for (int s = 128; s > 0; s >>= 1) {
        if (threadIdx.x < s) red[threadIdx.x] += red[threadIdx.x + s];
        __syncthreads();
    }
    float lse = logf(red[0]) + mx;
    for (int i = threadIdx.x; i < C; i += blockDim.x) p[i] = p[i] - lse;
}

// ---------------- orchestration ----------------

static void launch_gemm_simple(const float* A, const float* B, float* C,
                               int Mp, int Np, int K, int lda, int ldb, int ldc,
                               int Ms, int Ns, const float* bias, int act, hipStream_t stream) {
    int tiles = (Mp >> 5) * (Np >> 5);
    int blocks = cdiv(tiles, 8);
    gemm_wmma_f32<<<blocks, 256, 0, stream>>>(A, B, C, Mp, Np, K, lda, ldb, ldc,
                                              Ms, Ns, bias, act);
}

static void launch_gemm_lds(const float* A, const float* B, float* C,
                            int Mp, int Np, int K, int lda, int ldb, int ldc,
                            int Ms, int Ns, const float* bias, int act, hipStream_t stream) {
    int tnB = Np >> 7;
    int blocks = (Mp >> 7) * tnB;
    gemm_lds_wmma_f32<<<blocks, 256, 0, stream>>>(A, B, C, K, lda, ldb, ldc, tnB,
                                                  Ms, Ns, bias, act);
}

extern "C" void kernel_launch(void* const* d_in, const int* in_sizes, int n_in,
                              void* d_out, int out_size, void* d_ws, size_t ws_size,
                              hipStream_t stream) {
    static const int CH[7] = {1, 32, 64, 64, 64, 64, 64};

    const float* x0      = (const float*)d_in[0];
    const int*   eidx    = (const int*)d_in[1];
    const float* pseudo  = (const float*)d_in[2];
    const float* fc1_w   = (const float*)d_in[21];
    const float* fc1_b   = (const float*)d_in[22];
    const float* fc2_w   = (const float*)d_in[23];
    const float* fc2_b   = (const float*)d_in[24];
    const int* src = eidx;
    const int* dst = eidx + EE;

    float* ws = (float*)d_ws;
    size_t off = 0;
    float* basis = ws + off; off += (size_t)EE * 8;
    int*   flat  = (int*)(ws + off); off += (size_t)EE * 8;
    float* deg   = ws + off; off += NN;
    float* x0p   = ws + off; off += (size_t)NP * 4;
    float* xa    = ws + off; off += (size_t)NP * 64;
    float* xb    = ws + off; off += (size_t)NP * 64;
    float* agg   = ws + off; off += (size_t)NN * 64;
    float* rootb = ws + off; off += (size_t)NP * 64;
    float* rpad  = ws + off; off += (size_t)4 * 32;
    float* wt    = ws + off; off += (size_t)64 * LDH64;
    float* hch   = ws + off; off += (size_t)NP * LDH64;
    float* fc1o  = ws + off; off += (size_t)NP * 256;
    float* fc2wp = ws + off; off += (size_t)256 * FC2P;
    (void)ws_size; (void)in_sizes; (void)n_in; (void)out_size;

    basis_kernel<<<cdiv(EE, 256), 256, 0, stream>>>(pseudo, basis, flat);
    zero_kernel<<<cdiv(NN, 256), 256, 0, stream>>>(deg, NN);
    deg_kernel<<<cdiv(EE, 256), 256, 0, stream>>>(dst, deg);
    padmat_kernel<<<cdiv(NP * 4, 256), 256, 0, stream>>>(x0, x0p, NN, 1, 1, NP, 4, 4);
    padmat_kernel<<<cdiv(4 * 32, 256), 256, 0, stream>>>((const float*)d_in[4], rpad, 1, 32, 32, 4, 32, 32);
    padmat_kernel<<<cdiv(256 * FC2P, 256), 256, 0, stream>>>(fc2_w, fc2wp, 256, NN, NN, 256, FC2P, FC2P);
    zero_kernel<<<cdiv(NP * 64, 256), 256, 0, stream>>>(xa, NP * 64);
    zero_kernel<<<cdiv(NP * 64, 256), 256, 0, stream>>>(xb, NP * 64);

    const float* xcur = x0p;
    int ldx = 4;
    float* bufs[2] = {xa, xb};

    for (int l = 0; l < 6; l++) {
        const int cin = CH[l], cout = CH[l + 1];
        const int kpad = (cin < 4) ? 4 : cin;
        const float* Wc = (const float*)d_in[3 + 3 * l];
        const float* R  = (l == 0) ? rpad : (const float*)d_in[4 + 3 * l];
        const float* Bb = (const float*)d_in[5 + 3 * l];
        const int ldh = (l == 0) ? LDH32 : LDH64;

        zero_kernel<<<cdiv(NN * cout, 256), 256, 0, stream>>>(agg, NN * cout);

        // root term (small GEMM -> simple path)
        launch_gemm_simple(xcur, R, rootb, NP, cout, kpad, ldx, cout, cout,
                           NP, cout, nullptr, 0, stream);

        for (int c = 0; c < NCHUNK; c++) {
            const int k0 = c * KCH;
            pack_kernel<<<cdiv(kpad * ldh, 256), 256, 0, stream>>>(
                Wc, wt, cin, cout, k0, KCH, kpad, ldh);
            if (l == 0) {
                launch_gemm_simple(xcur, wt, hch, NP, ldh, kpad, ldx, ldh, ldh,
                                   NP, ldh, nullptr, 0, stream);
            } else {
                launch_gemm_lds(xcur, wt, hch, NP, ldh, kpad, ldx, ldh, ldh,
                                NP, ldh, nullptr, 0, stream);
            }
            edge_kernel<<<cdiv(EE * cout, 256), 256, 0, stream>>>(
                hch, src, dst, basis, flat, agg, cout, k0, k0 + KCH, ldh);
        }

        node_kernel<<<cdiv(NN * cout, 256), 256, 0, stream>>>(
            agg, rootb, Bb, deg, bufs[l & 1], cout, 64, cout);
        xcur = bufs[l & 1];
        ldx = 64;
    }

    launch_gemm_lds(xcur, fc1_w, fc1o, NP, 256, 64, 64, 256, 256,
                    NP, 256, fc1_b, 1, stream);
    launch_gemm_lds(fc1o, fc2wp, (float*)d_out, NP, FC2P, 256, 256, FC2P, NN,
                    NN, NN, fc2_b, 0, stream);
    logsoftmax_kernel<<<NN, 256, 0, stream>>>((float*)d_out, NN);
}